// VoxelPooling_57397942944129
// MI455X (gfx1250) — compile-verified
//
#include <hip/hip_runtime.h>
#include <hip/hip_bf16.h>

// Problem dims (fixed by the reference harness)
#define BB    4
#define NN    4
#define CC    80
#define HH    16
#define WW    44
#define DD    64
#define XX    128
#define YY    128
#define ZZ    64
#define XY    (XX * YY)        // 16384
#define DT    (DD / 16)        // 4 depth tiles
#define CT    5                // 5 channel tiles (80/16)
#define NTILE (BB * NN * HH * WW * DT)   // 45056 waves
#define OUT_ELEMS (BB * CC * XY)         // 5242880 floats

typedef float v2f __attribute__((ext_vector_type(2)));
typedef float v8f __attribute__((ext_vector_type(8)));

__device__ __forceinline__ void inv3x3(const float m[9], float o[9]) {
    float a = m[0], b = m[1], c = m[2];
    float d = m[3], e = m[4], f = m[5];
    float g = m[6], h = m[7], i = m[8];
    float A =  (e * i - f * h);
    float B = -(d * i - f * g);
    float C =  (d * h - e * g);
    float det = a * A + b * B + c * C;
    float inv = 1.0f / det;
    o[0] = A * inv;               o[1] = -(b * i - c * h) * inv;  o[2] =  (b * f - c * e) * inv;
    o[3] = B * inv;               o[4] =  (a * i - c * g) * inv;  o[5] = -(a * f - c * d) * inv;
    o[6] = C * inv;               o[7] = -(a * h - b * g) * inv;  o[8] =  (a * e - b * d) * inv;
}

// Invert the 16 intrinsic (3x3) and extrinsic (4x4, [R t; 0 1]) matrices.
__global__ void vp_prep_kernel(const float* __restrict__ intr,
                               const float* __restrict__ extr,
                               float* __restrict__ kinv,
                               float* __restrict__ einv) {
    int t = threadIdx.x;
    if (t >= BB * NN) return;
    // K^-1
    float K[9], Ki[9];
    #pragma unroll
    for (int i = 0; i < 9; ++i) K[i] = intr[t * 9 + i];
    inv3x3(K, Ki);
    #pragma unroll
    for (int i = 0; i < 9; ++i) kinv[t * 9 + i] = Ki[i];
    // E^-1 = [Rinv, -Rinv*t; 0 0 0 1]
    const float* E = extr + t * 16;
    float R[9] = {E[0], E[1], E[2], E[4], E[5], E[6], E[8], E[9], E[10]};
    float tv[3] = {E[3], E[7], E[11]};
    float Ri[9];
    inv3x3(R, Ri);
    float* ei = einv + t * 16;
    #pragma unroll
    for (int r = 0; r < 3; ++r) {
        ei[r * 4 + 0] = Ri[r * 3 + 0];
        ei[r * 4 + 1] = Ri[r * 3 + 1];
        ei[r * 4 + 2] = Ri[r * 3 + 2];
        ei[r * 4 + 3] = -(Ri[r * 3 + 0] * tv[0] + Ri[r * 3 + 1] * tv[1] + Ri[r * 3 + 2] * tv[2]);
    }
    ei[12] = 0.0f; ei[13] = 0.0f; ei[14] = 0.0f; ei[15] = 1.0f;
}

__global__ void vp_zero_kernel(float4* __restrict__ out, int n4) {
    int i = blockIdx.x * 256 + threadIdx.x;
    if (i < n4) out[i] = make_float4(0.0f, 0.0f, 0.0f, 0.0f);
}

// One wave per (b, n, h, w, 16-depth tile).  Weight tile (16 depths x 16
// channels) is one rank-1 outer product formed by V_WMMA_F32_16X16X4_F32;
// voxel indices are computed one-row-per-lane and redistributed with shfl.
__global__ __launch_bounds__(256) void vp_splat_kernel(
        const float* __restrict__ ctx,   // [B,C,H,W]
        const float* __restrict__ dp,    // [N,D,H,W]
        const float* __restrict__ kinv,  // [B*N,9]
        const float* __restrict__ einv,  // [B*N,16]
        float* __restrict__ out) {       // [B,C,X,Y]
    const int lane = threadIdx.x & 31;
    int t = blockIdx.x * 8 + (threadIdx.x >> 5);   // tile id, < NTILE exactly
    const int dt = t & 3;          t >>= 2;
    const int w  = t % WW;         t /= WW;
    const int h  = t & (HH - 1);   t >>= 4;
    const int n  = t & 3;
    const int b  = t >> 2;

    const float* Ki = kinv + (b * NN + n) * 9;
    const float* Ei = einv + (b * NN + n) * 16;

    // cam = [w, h, 1] @ K^-1   (row-vector)
    const float px = (float)w, py = (float)h;
    const float cam0 = px * Ki[0] + py * Ki[3] + Ki[6];
    const float cam1 = px * Ki[1] + py * Ki[4] + Ki[7];
    const float cam2 = px * Ki[2] + py * Ki[5] + Ki[8];
    // world_j(depth) = depth * a_j + c_j   (c_j = Einv[3][j] = 0 analytically)
    const float a0 = cam0 * Ei[0] + cam1 * Ei[4] + cam2 * Ei[8];
    const float a1 = cam0 * Ei[1] + cam1 * Ei[5] + cam2 * Ei[9];
    const float a2 = cam0 * Ei[2] + cam1 * Ei[6] + cam2 * Ei[10];
    const float c0 = Ei[12], c1 = Ei[13], c2 = Ei[14];

    // Per-lane voxel index for row mrow of this depth tile
    // lanes 0-15 cover rows 0-7 (dup), lanes 16-31 cover rows 8-15 (dup)
    const int mrow = ((lane >> 4) << 3) | (lane & 7);
    const int mg = dt * 16 + mrow;
    const float depth = 1.0f + (float)mg * (59.0f / 63.0f);   // linspace(1,60,64)
    const float wx = depth * a0 + c0;
    const float wy = depth * a1 + c1;
    const float wz = depth * a2 + c2;
    const int xi = (int)((wx + 51.2f) / 0.8f);   // trunc toward zero, == .astype(int32)
    const int yi = (int)((wy + 51.2f) / 0.8f);
    const int zi = (int)wz;
    const bool valid = (xi >= 0) & (xi < XX) & (yi >= 0) & (yi < YY) & (zi >= 0) & (zi < ZZ);
    const int lin = valid ? (xi * YY + yi) : -1;

    // Row r needed by this lane lives in lane (lane&16)|r
    int linr[8];
    #pragma unroll
    for (int r = 0; r < 8; ++r) linr[r] = __shfl(lin, (lane & 16) | r, 32);

    // A matrix (16x4 f32): column K=0 = depth probs for the 16 rows.
    // Layout: lanes 0-15 = M=0..15, VGPR0 = K=0 -> only lanes<16 x-component.
    float aval = 0.0f;
    if (lane < 16)
        aval = dp[((n * DD + dt * 16 + lane) * HH + h) * WW + w];
    v2f Am = {aval, 0.0f};

    #pragma unroll
    for (int ct = 0; ct < CT; ++ct) {
        const int c = ct * 16 + (lane & 15);
        // B matrix (4x16 f32): row K=0 = context channels (lanes 0-15, VGPR0).
        float bval = 0.0f;
        if (lane < 16)
            bval = ctx[((b * CC + c) * HH + h) * WW + w];
        v2f Bm = {bval, 0.0f};
        v8f acc = {};
        // D = A x B : 16x16 outer product dp[m] * ctx[c]
        acc = __builtin_amdgcn_wmma_f32_16x16x4_f32(
            /*neg_a=*/false, Am, /*neg_b=*/false, Bm,
            /*c_mod=*/(short)0, acc, /*reuse_a=*/false, /*reuse_b=*/false);

        // D layout: VGPR r -> M = r (lanes 0-15) / r+8 (lanes 16-31), N = lane%16
        float* outc = out + (size_t)(b * CC + c) * XY;
        #pragma unroll
        for (int r = 0; r < 8; ++r) {
            if (linr[r] >= 0) {
                float v = acc[r];
                (void)__hip_atomic_fetch_add(outc + linr[r], v,
                                             __ATOMIC_RELAXED, __HIP_MEMORY_SCOPE_AGENT);
            }
        }
    }
}

extern "C" void kernel_launch(void* const* d_in, const int* in_sizes, int n_in,
                              void* d_out, int out_size, void* d_ws, size_t ws_size,
                              hipStream_t stream) {
    const float* ctx  = (const float*)d_in[0];   // context   [4,80,16,44]
    const float* dp   = (const float*)d_in[1];   // depth_prob[4,64,16,44]
    const float* intr = (const float*)d_in[2];   // [4,4,3,3]
    const float* extr = (const float*)d_in[3];   // [4,4,4,4]
    float* out = (float*)d_out;                  // [4,80,128,128]

    float* kinv = (float*)d_ws;                  // 144 floats
    float* einv = kinv + 144;                    // 256 floats

    vp_prep_kernel<<<1, 32, 0, stream>>>(intr, extr, kinv, einv);
    vp_zero_kernel<<<OUT_ELEMS / 4 / 256, 256, 0, stream>>>((float4*)out, OUT_ELEMS / 4);
    vp_splat_kernel<<<NTILE / 8, 256, 0, stream>>>(ctx, dp, kinv, einv, out);
}